// GeoBleuSinkhornLoss_45853070852124
// MI455X (gfx1250) — compile-verified
//
#include <hip/hip_runtime.h>
#include <hip/hip_bf16.h>
#include <math.h>

typedef __attribute__((ext_vector_type(2))) float v2f;
typedef __attribute__((ext_vector_type(8))) float v8f;

#define VV 40000
#define TT 64
#define BB 32
#define NWIN 49

// ---------------- Kernel 1: online softmax stats per row (b,t) ----------------
__global__ void k_row_stats(const float* __restrict__ logits,
                            float* __restrict__ rowmax,
                            float* __restrict__ rowsum) {
    const int row = blockIdx.x;                 // 0..2047
    const float4* rp4 = (const float4*)(logits + (size_t)row * VV);
    float m = -3.4e38f, s = 0.0f;
    for (int i = threadIdx.x; i < VV / 4; i += 256) {
        float4 v = rp4[i];
        float xm = fmaxf(fmaxf(v.x, v.y), fmaxf(v.z, v.w));
        float nm = fmaxf(m, xm);
        s = s * expf(m - nm)
          + expf(v.x - nm) + expf(v.y - nm) + expf(v.z - nm) + expf(v.w - nm);
        m = nm;
    }
    __shared__ float sm[256], ss[256];
    sm[threadIdx.x] = m; ss[threadIdx.x] = s;
    __syncthreads();
    for (int off = 128; off > 0; off >>= 1) {
        if (threadIdx.x < off) {
            float m1 = sm[threadIdx.x], s1 = ss[threadIdx.x];
            float m2 = sm[threadIdx.x + off], s2 = ss[threadIdx.x + off];
            float nm = fmaxf(m1, m2);
            sm[threadIdx.x] = nm;
            ss[threadIdx.x] = s1 * expf(m1 - nm) + s2 * expf(m2 - nm);
        }
        __syncthreads();
    }
    if (threadIdx.x == 0) { rowmax[row] = sm[0]; rowsum[row] = ss[0]; }
}

// ---------------- Kernel 2: U[b,i,j] = sum_k p[b,i,nb[tgt,k]] * wnorm ----------------
__global__ void k_unigram(const float* __restrict__ logits,
                          const int* __restrict__ tgt,
                          const float* __restrict__ kw,
                          const int* __restrict__ nb,
                          const unsigned char* __restrict__ valid,
                          const float* __restrict__ rowmax,
                          const float* __restrict__ rowsum,
                          float* __restrict__ U) {
    const int bi = blockIdx.x;                  // b*64 + i
    const int b  = bi >> 6;
    const int j  = threadIdx.x;                 // 0..63
    const int t  = tgt[b * TT + j];
    const float m = rowmax[bi];
    const float inv_s = 1.0f / rowsum[bi];
    const float* rp = logits + (size_t)bi * VV;
    float acc = 0.0f, wsum = 0.0f;
    for (int k = 0; k < NWIN; ++k) {
        const float w  = valid[t * NWIN + k] ? kw[k] : 0.0f;
        const int   id = nb[t * NWIN + k];
        const float p  = expf(rp[id] - m) * inv_s;
        acc  += p * w;
        wsum += w;
    }
    const float u = acc / fmaxf(wsum, 1e-12f);
    U[(size_t)bi * TT + j] = fmaxf(u, 1e-12f);
}

// ---------------- Kernel 3: per (b,n) Sinkhorn with f32 WMMA matvecs ----------------
// K and K^T A-fragments are hoisted into VGPRs (iteration-invariant); the
// 30-iteration loop only reads the current a/b vector broadcasts from LDS.
__global__ void k_sinkhorn(const float* __restrict__ U,
                           const float* __restrict__ weights,
                           float* __restrict__ part) {
    const int n_idx = blockIdx.x % 5;
    const int b     = blockIdx.x / 5;
    const int n     = n_idx + 1;
    const int I     = TT - n + 1;

    __shared__ float Smat[64 * 64];
    __shared__ float Kmat[64 * 64];
    __shared__ float KT  [64 * 64];
    __shared__ float avec[64], bvec[64], tvec[64];
    __shared__ float red[128];

    const float* Ub = U + (size_t)b * TT * TT;
    const int tid = threadIdx.x;                // 0..127

    for (int idx = tid; idx < 4096; idx += 128) {
        const int i = idx >> 6, j = idx & 63;
        float s = 0.0f, kk = 0.0f;
        if (i < I && j < I) {
            s = 1.0f;
            for (int k = 0; k < n; ++k) s *= Ub[(i + k) * TT + (j + k)];
            s  = fmaxf(s, 1e-12f);
            kk = fmaxf(expf(s * 10.0f), 1e-30f);   // 1/EPS = 10
        }
        Smat[idx] = s;
        Kmat[idx] = kk;
        KT[j * 64 + i] = kk;
    }
    if (tid < 64) { avec[tid] = 1.0f; bvec[tid] = 1.0f; }
    __syncthreads();

    const int wave = tid >> 5;
    const int lane = tid & 31;
    const int half = lane >> 4;       // K half selection (lanes 0-15 vs 16-31)
    const int mrow = lane & 15;       // M index within tile
    const int r0   = wave * 16;       // this wave's row block
    const float muI = 1.0f / (float)I;

    // Hoist iteration-invariant A-matrix fragments (K and K^T) into registers:
    // 16 chunks x v2f each = 32 VGPRs per matrix.
    v2f ak[16], akt[16];
#pragma unroll
    for (int kc = 0; kc < 16; ++kc) {
        const int kb = kc * 4 + half * 2;
        ak[kc].x  = Kmat[(r0 + mrow) * 64 + kb];
        ak[kc].y  = Kmat[(r0 + mrow) * 64 + kb + 1];
        akt[kc].x = KT  [(r0 + mrow) * 64 + kb];
        akt[kc].y = KT  [(r0 + mrow) * 64 + kb + 1];
    }

    for (int it = 0; it < 30; ++it) {
        // t = K @ b  (B-operand broadcasts bvec across all 16 columns)
        {
            v8f c = {0.f,0.f,0.f,0.f,0.f,0.f,0.f,0.f};
#pragma unroll
            for (int kc = 0; kc < 16; ++kc) {
                const int kb = kc * 4 + half * 2;
                v2f bf;
                bf.x = bvec[kb];
                bf.y = bvec[kb + 1];
                c = __builtin_amdgcn_wmma_f32_16x16x4_f32(false, ak[kc], false, bf,
                                                          (short)0, c, false, false);
            }
            if (lane == 0)  { for (int g = 0; g < 8; ++g) tvec[r0 + g]     = c[g]; }
            if (lane == 16) { for (int g = 0; g < 8; ++g) tvec[r0 + 8 + g] = c[g]; }
        }
        __syncthreads();
        if (tid < 64) {
            const float mu = (tid < I) ? muI : 0.0f;
            avec[tid] = mu / fmaxf(tvec[tid], 1e-30f);
        }
        __syncthreads();
        // u = K^T @ a
        {
            v8f c = {0.f,0.f,0.f,0.f,0.f,0.f,0.f,0.f};
#pragma unroll
            for (int kc = 0; kc < 16; ++kc) {
                const int kb = kc * 4 + half * 2;
                v2f bf;
                bf.x = avec[kb];
                bf.y = avec[kb + 1];
                c = __builtin_amdgcn_wmma_f32_16x16x4_f32(false, akt[kc], false, bf,
                                                          (short)0, c, false, false);
            }
            if (lane == 0)  { for (int g = 0; g < 8; ++g) tvec[r0 + g]     = c[g]; }
            if (lane == 16) { for (int g = 0; g < 8; ++g) tvec[r0 + 8 + g] = c[g]; }
        }
        __syncthreads();
        if (tid < 64) {
            const float nu = (tid < I) ? muI : 0.0f;
            bvec[tid] = nu / fmaxf(tvec[tid], 1e-30f);
        }
        __syncthreads();
    }

    // q_n = sum_ij a_i * K_ij * b_j * S_ij
    float acc = 0.0f;
    for (int idx = tid; idx < 4096; idx += 128) {
        const int i = idx >> 6, j = idx & 63;
        acc += avec[i] * Kmat[idx] * bvec[j] * Smat[idx];
    }
    red[tid] = acc;
    __syncthreads();
    for (int off = 64; off > 0; off >>= 1) {
        if (tid < off) red[tid] += red[tid + off];
        __syncthreads();
    }
    if (tid == 0) {
        const float q = fmaxf(red[0], 1e-12f);
        part[b * 5 + n_idx] = weights[n_idx] * logf(q);
    }
}

// ---------------- Kernel 4: loss = -(1/B) * sum(part) ----------------
__global__ void k_final(const float* __restrict__ part, float* __restrict__ out) {
    __shared__ float red[256];
    const int t = threadIdx.x;
    red[t] = (t < BB * 5) ? part[t] : 0.0f;
    __syncthreads();
    for (int off = 128; off > 0; off >>= 1) {
        if (t < off) red[t] += red[t + off];
        __syncthreads();
    }
    if (t == 0) out[0] = -red[0] / (float)BB;
}

extern "C" void kernel_launch(void* const* d_in, const int* in_sizes, int n_in,
                              void* d_out, int out_size, void* d_ws, size_t ws_size,
                              hipStream_t stream) {
    const float*         logits  = (const float*)d_in[0];          // (32,64,40000)
    const int*           tgt     = (const int*)d_in[1];            // (32,64)
    const float*         kw      = (const float*)d_in[2];          // (49,)
    const int*           nb      = (const int*)d_in[3];            // (40000,49)
    const unsigned char* valid   = (const unsigned char*)d_in[4];  // (40000,49) bool
    const float*         weights = (const float*)d_in[5];          // (5,)
    float* out = (float*)d_out;

    float* ws     = (float*)d_ws;
    float* rowmax = ws;                        // 2048
    float* rowsum = ws + 2048;                 // 2048
    float* U      = ws + 4096;                 // 32*64*64 = 131072
    float* part   = ws + 4096 + 131072;        // 160

    k_row_stats<<<BB * TT, 256, 0, stream>>>(logits, rowmax, rowsum);
    k_unigram  <<<BB * TT, 64,  0, stream>>>(logits, tgt, kw, nb, valid,
                                             rowmax, rowsum, U);
    k_sinkhorn <<<BB * 5, 128,  0, stream>>>(U, weights, part);
    k_final    <<<1, 256,       0, stream>>>(part, out);
}